// BoundaryPredictor2_45045617001133
// MI455X (gfx1250) — compile-verified
//
#include <hip/hip_runtime.h>
#include <hip/hip_bf16.h>
#include <stdint.h>

typedef float v2f __attribute__((ext_vector_type(2)));
typedef float v8f __attribute__((ext_vector_type(8)));
typedef uint32_t u32x4 __attribute__((ext_vector_type(4)));
typedef int i32x4 __attribute__((ext_vector_type(4)));
typedef int i32x8 __attribute__((ext_vector_type(8)));

#define B_  8
#define L_  2048
#define D_  1024
#define EPS_ 1.1920929e-07f
#define NROWS_ (B_ * L_)          // 16384 flattened (b,l) rows

// LDS A-tile row stride in floats: 64 data + 1 pad dword (TDM pad_interval=5 ->
// pad after every 64 dwords, pad_amount=0 -> 1 dword)
#define LDS_STRIDE 65
#define TILE_BYTES (16 * LDS_STRIDE * 4)

// ---------------------------------------------------------------------------
// helpers
// ---------------------------------------------------------------------------
__device__ __forceinline__ float blockReduceSum256(float v, float* sbuf) {
  int t = threadIdx.x;
  sbuf[t] = v;
  __syncthreads();
  for (int s = 128; s > 0; s >>= 1) {
    if (t < s) sbuf[t] += sbuf[t + s];
    __syncthreads();
  }
  float r = sbuf[0];
  __syncthreads();
  return r;
}

// threefry2x32 with key (0, 42)  == jax.random.key(42)
__device__ __forceinline__ void threefry2x32_42(uint32_t x0, uint32_t x1,
                                                uint32_t& o0, uint32_t& o1) {
  const uint32_t ks0 = 0u, ks1 = 42u;
  const uint32_t ks2 = ks0 ^ ks1 ^ 0x1BD11BDAu;
  uint32_t ks[3] = {ks0, ks1, ks2};
  const int rotA[4] = {13, 15, 26, 6};
  const int rotB[4] = {17, 29, 16, 24};
  x0 += ks[0];
  x1 += ks[1];
#pragma unroll
  for (int i = 0; i < 5; ++i) {
    const int* rot = ((i & 1) == 0) ? rotA : rotB;
#pragma unroll
    for (int j = 0; j < 4; ++j) {
      x0 += x1;
      x1 = (x1 << rot[j]) | (x1 >> (32 - rot[j]));
      x1 ^= x0;
    }
    x0 += ks[(i + 1) % 3];
    x1 += ks[(i + 2) % 3] + (uint32_t)(i + 1);
  }
  o0 = x0;
  o1 = x1;
}

// ---------------------------------------------------------------------------
// K0: transpose 1024x1024:  WT[d][j] = W[j][d]
// ---------------------------------------------------------------------------
__global__ void transpose1024_kernel(const float* __restrict__ W,
                                     float* __restrict__ WT) {
  int i = blockIdx.x * 256 + threadIdx.x;   // 1M elements
  int j = i & (D_ - 1);
  int d = i >> 10;
  WT[i] = W[j * D_ + d];
}

// ---------------------------------------------------------------------------
// K1: per-row reciprocal L2 norm (F.normalize eps=1e-12)
// ---------------------------------------------------------------------------
__global__ void rnorm_kernel(const float* __restrict__ H,
                             float* __restrict__ rnorm) {
  __shared__ float sbuf[256];
  int row = blockIdx.x;
  const float* p = H + (size_t)row * D_;
  float ss = 0.0f;
  for (int c = threadIdx.x; c < D_; c += 256) {
    float v = p[c];
    ss += v * v;
  }
  float tot = blockReduceSum256(ss, sbuf);
  if (threadIdx.x == 0) {
    float nrm = sqrtf(tot);
    rnorm[row] = 1.0f / fmaxf(nrm, 1e-12f);
  }
}

// ---------------------------------------------------------------------------
// TDM: issue a 16-row x 64-float tile load H[m0+y][kc+x] -> LDS (stride 65),
// using the Tensor Data Mover (D# built per CDNA5 ISA).
// Wave-level operation: call from one wave only; tracked by TENSORcnt.
// ---------------------------------------------------------------------------
__device__ __forceinline__ void tdm_load_a_tile(const float* H, int m0, int kc,
                                                uint32_t lds_addr) {
  uint64_t gaddr = (uint64_t)(uintptr_t)H +
                   (((uint64_t)m0 * D_ + (uint64_t)kc) << 2);

  // ---- D# group 0 (4 dwords): count=1, lds_addr, global_addr, type=2
  u32x4 g0;
  g0.x = 1u;                                   // count=1, is_restore=0, no gather
  g0.y = lds_addr;                             // LDS byte address
  g0.z = (uint32_t)(gaddr & 0xFFFFFFFFu);      // global_addr[31:0]
  g0.w = (uint32_t)((gaddr >> 32) & 0x01FFFFFFu) | (2u << 30);  // [56:32] | type=2

  // ---- D# group 1 (8 dwords)
  // dword0: wg_mask=0 | data_size=2(4B)<<16 | pad_enable<<20 | pad_interval=5<<22
  i32x8 g1;
  g1[0] = (int)((2u << 16) | (1u << 20) | (5u << 22));
  g1[1] = (int)(((uint32_t)(D_ & 0xFFFF)) << 16);       // tensor_dim0[15:0]=1024
  g1[2] = (int)((((uint32_t)NROWS_) & 0xFFFFu) << 16);  // dim0[31:16]=0 | tensor_dim1[15:0]
  g1[3] = (int)(((uint32_t)64) << 16);                  // tensor_dim1[31:16]=0 | tile_dim0=64
  g1[4] = 16;                                           // tile_dim1=16, tile_dim2=0
  g1[5] = D_;                                           // tensor_dim0_stride[31:0]=1024
  g1[6] = 0;                                            // stride[47:32]=0 | dim1_stride lo
  g1[7] = 0;                                            // dim1_stride hi

  i32x4 gz4 = {0, 0, 0, 0};
  i32x8 gz8 = {0, 0, 0, 0, 0, 0, 0, 0};

#if __has_builtin(__builtin_amdgcn_tensor_load_to_lds)
  // 6-arg form (amdgpu-toolchain / clang-23): (v4u, v8i, v4i, v4i, v8i, cpol)
  __builtin_amdgcn_tensor_load_to_lds(g0, g1, gz4, gz4, gz8, 0);
#else
  (void)g0; (void)g1; (void)gz4; (void)gz8;
#endif
}

// ---------------------------------------------------------------------------
// K2: WMMA fp32 GEMM.  OUT[m,n] = sum_d H[m,d] * WT[d,n]   (un-normalized;
//     the per-row 1/||h|| factors are applied later in the cos-sim kernel)
//     M = 16384, N = 1024, K = 1024.
//     Block = 128 threads (4 waves). Each wave: one 16(M) x 64(N) strip
//     (4 WMMA tiles). A tile double-buffered in LDS via the Tensor Data
//     Mover, overlapped with WMMA compute.
// ---------------------------------------------------------------------------
__global__ void gemm_wmma_kernel(const float* __restrict__ H,
                                 const float* __restrict__ WT,
                                 float* __restrict__ OUT) {
  __shared__ float lds_a[2 * 16 * LDS_STRIDE];

  const int m0   = blockIdx.x * 16;
  const int wave = threadIdx.x >> 5;
  const int lane = threadIdx.x & 31;
  const int n0   = blockIdx.y * 256 + wave * 64;

  v8f acc0 = {0,0,0,0,0,0,0,0};
  v8f acc1 = {0,0,0,0,0,0,0,0};
  v8f acc2 = {0,0,0,0,0,0,0,0};
  v8f acc3 = {0,0,0,0,0,0,0,0};

  const int arow = lane & 15;
  const int kofs = (lane >> 4) * 2;     // A VGPR0 holds K, K+2 per ISA layout
  const int bn   = lane & 15;
  const int bko  = (lane >> 4) * 2;     // B symmetric: rows K / K+2 in v0

  const uint32_t lds_base = (uint32_t)(uintptr_t)&lds_a[0];
  const int NCHUNK = D_ / 64;           // 16

  // prologue: TDM fetch of chunk 0 into buffer 0
  if (wave == 0) {
    tdm_load_a_tile(H, m0, 0, lds_base);
  }

  for (int i = 0; i < NCHUNK; ++i) {
    __syncthreads();   // all waves done reading the buffer we are about to refill
    if (wave == 0) {
      if (i + 1 < NCHUNK) {
        tdm_load_a_tile(H, m0, (i + 1) * 64,
                        lds_base + (uint32_t)(((i + 1) & 1) * TILE_BYTES));
        __builtin_amdgcn_s_wait_tensorcnt(1);   // chunk i landed, i+1 in flight
      } else {
        __builtin_amdgcn_s_wait_tensorcnt(0);
      }
    }
    __syncthreads();   // publish chunk i

    const float* cur = &lds_a[(i & 1) * 16 * LDS_STRIDE];
    const int kc = i * 64;

    // prefetch next B chunk into L2 (global_prefetch_b8)
    if (i + 1 < NCHUNK)
      __builtin_prefetch(WT + (size_t)(kc + 64) * D_ + n0 + lane, 0, 1);

#pragma unroll
    for (int kk = 0; kk < 64; kk += 4) {
      v2f a;
      a.x = cur[arow * LDS_STRIDE + kk + kofs];
      a.y = cur[arow * LDS_STRIDE + kk + kofs + 1];
      const int kg = kc + kk + bko;
      const float* wt0 = WT + (size_t)kg * D_;

      v2f b0, b1, b2, b3;
      b0.x = wt0[n0 +  0 + bn]; b0.y = wt0[D_ + n0 +  0 + bn];
      b1.x = wt0[n0 + 16 + bn]; b1.y = wt0[D_ + n0 + 16 + bn];
      b2.x = wt0[n0 + 32 + bn]; b2.y = wt0[D_ + n0 + 32 + bn];
      b3.x = wt0[n0 + 48 + bn]; b3.y = wt0[D_ + n0 + 48 + bn];

      acc0 = __builtin_amdgcn_wmma_f32_16x16x4_f32(false, a, false, b0, (short)0, acc0, false, false);
      acc1 = __builtin_amdgcn_wmma_f32_16x16x4_f32(false, a, false, b1, (short)0, acc1, false, false);
      acc2 = __builtin_amdgcn_wmma_f32_16x16x4_f32(false, a, false, b2, (short)0, acc2, false, false);
      acc3 = __builtin_amdgcn_wmma_f32_16x16x4_f32(false, a, false, b3, (short)0, acc3, false, false);
    }
  }

  // C/D layout: VGPR r -> M = r (lanes 0-15) / r+8 (lanes 16-31), N = lane&15
  const int half = lane >> 4;
  const int col  = lane & 15;
#pragma unroll
  for (int r = 0; r < 8; ++r) {
    size_t row = (size_t)(m0 + r + half * 8) * D_;
    OUT[row + n0 +  0 + col] = acc0[r];
    OUT[row + n0 + 16 + col] = acc1[r];
    OUT[row + n0 + 32 + col] = acc2[r];
    OUT[row + n0 + 48 + col] = acc3[r];
  }
}

// ---------------------------------------------------------------------------
// K3: cos_sim[b,l] = rnorm[b,l]*rnorm[b,l+1]*dot(qraw[b,l], kraw[b,l+1])
// ---------------------------------------------------------------------------
__global__ void cossim_kernel(const float* __restrict__ q,
                              const float* __restrict__ k,
                              const float* __restrict__ rnorm,
                              float* __restrict__ cosbuf) {
  int gw   = blockIdx.x * 8 + (threadIdx.x >> 5);
  int lane = threadIdx.x & 31;
  if (gw >= B_ * (L_ - 1)) return;
  int b = gw / (L_ - 1);
  int l = gw % (L_ - 1);
  size_t row = (size_t)b * L_ + l;
  const float* qr = q + row * D_;
  const float* kr = k + (row + 1) * D_;
  float s = 0.0f;
  for (int c = lane; c < D_; c += 32) s += qr[c] * kr[c];
  for (int off = 16; off > 0; off >>= 1) s += __shfl_xor(s, off, 32);
  if (lane == 0)
    cosbuf[b * (L_ - 1) + l] = s * rnorm[row] * rnorm[row + 1];
}

// ---------------------------------------------------------------------------
// K4: per-batch boundary logic (probs -> relaxed-bernoulli -> hard -> scan)
// ---------------------------------------------------------------------------
__global__ void boundary_kernel(const float* __restrict__ mask,
                                const float* __restrict__ cosbuf,
                                float* __restrict__ hardWS,
                                int* __restrict__ starts,   // [B][L+1]
                                int* __restrict__ nsegA,
                                float* __restrict__ kbA,
                                float* __restrict__ nbA,
                                float* __restrict__ samOut) {
  __shared__ float sred[256];
  __shared__ int   sscan[256];
  __shared__ int   s_total;

  const int b = blockIdx.x;
  const int t = threadIdx.x;

  float m[8];
  float msum = 0.0f;
#pragma unroll
  for (int j = 0; j < 8; ++j) {
    m[j] = mask[b * L_ + t * 8 + j];
    msum += m[j];
  }
  float n = blockReduceSum256(msum, sred);
  int len = (int)(n + 0.5f);

  int hv[8];
#pragma unroll
  for (int j = 0; j < 8; ++j) {
    int l = t * 8 + j;
    float p;
    if (l == 0) {
      p = 1.0f;
    } else {
      float cs = cosbuf[b * (L_ - 1) + l - 1];
      p = 0.5f * (1.0f - cs);
      p = fminf(fmaxf(p, 0.0f), 1.0f);
    }
    float pc = fminf(fmaxf(p, EPS_), 1.0f - EPS_);
    float logits = logf(pc) - log1pf(-pc);

    uint32_t idx = (uint32_t)(b * L_ + l);
    uint32_t o0, o1, bits;
    if (idx < 8192u) { threefry2x32_42(idx, idx + 8192u, o0, o1); bits = o0; }
    else             { threefry2x32_42(idx - 8192u, idx, o0, o1); bits = o1; }
    float f = __uint_as_float((bits >> 9) | 0x3F800000u) - 1.0f;
    float u = f * ((1.0f - EPS_) - EPS_) + EPS_;
    u = fmaxf(EPS_, u);
    float noise = logf(u) - log1pf(-u);

    float soft = 1.0f / (1.0f + expf(-(logits + noise)));
    int h = (soft > 0.5f) ? 1 : 0;
    if (m[j] == 0.0f) h = 0;                 // hard *= attention_mask
    if (len < L_ && l == len - 1) h = 1;     // forced last-real boundary
    hv[j] = h;
    hardWS[b * L_ + l] = (float)h;
  }

  // exclusive scan over 2048 hard bits
  int lpre[8];
  int run = 0;
#pragma unroll
  for (int j = 0; j < 8; ++j) { lpre[j] = run; run += hv[j]; }
  sscan[t] = run;
  __syncthreads();
  if (t == 0) {
    int acc = 0;
    for (int i = 0; i < 256; ++i) { int tmp = sscan[i]; sscan[i] = acc; acc += tmp; }
    s_total = acc;
  }
  __syncthreads();
  int base = sscan[t];
  int kb_i = s_total;

  // segment starts (seg increments by 1 exactly where hard[l-1]==1)
#pragma unroll
  for (int j = 0; j < 8; ++j) {
    int l = t * 8 + j;
    int s = base + lpre[j];
    if (l == 0) starts[b * (L_ + 1)] = 0;
    else if (hardWS[b * L_ + l - 1] > 0.5f) starts[b * (L_ + 1) + s] = l;
  }
  if (t == 255) {   // owns l = 2047
    int nseg = kb_i - hv[7] + 1;
    nsegA[b] = nseg;
    starts[b * (L_ + 1) + nseg] = L_;
  }
  if (t == 0) { kbA[b] = (float)kb_i; nbA[b] = n; }

  // sam row: ones for r < k_b
  for (int r = t; r < L_; r += 256)
    samOut[b * L_ + r] = (r < kb_i) ? 1.0f : 0.0f;
}

// ---------------------------------------------------------------------------
// K5: segment mean pooling. Block (b, s): mean of hidden rows in segment s.
// ---------------------------------------------------------------------------
__global__ void pool_kernel(const float* __restrict__ H,
                            const int* __restrict__ starts,
                            const int* __restrict__ nsegA,
                            float* __restrict__ pooled) {
  const int s = blockIdx.x;
  const int b = blockIdx.y;
  const int t = threadIdx.x;
  float* out = pooled + ((size_t)b * L_ + s) * D_;
  int nseg = nsegA[b];
  if (s >= nseg) {
    for (int c = t; c < D_; c += 256) out[c] = 0.0f;
    return;
  }
  int st = starts[b * (L_ + 1) + s];
  int en = starts[b * (L_ + 1) + s + 1];
  float scale = 1.0f / ((float)(en - st) + 1e-9f);
  for (int c = t; c < D_; c += 256) {
    float acc = 0.0f;
    for (int r = st; r < en; ++r)
      acc += H[((size_t)b * L_ + r) * D_ + c];
    out[c] = acc * scale;
  }
}

// ---------------------------------------------------------------------------
// K6: scalars -- binomial loss, num_boundaries, total_positions
// ---------------------------------------------------------------------------
__global__ void finalize_kernel(const float* __restrict__ kbA,
                                const float* __restrict__ nbA,
                                const int* __restrict__ tgt,
                                float* __restrict__ out3) {
  if (threadIdx.x != 0 || blockIdx.x != 0) return;
  float sumk = 0.0f, sumn = 0.0f, loss = 0.0f;
  for (int b = 0; b < B_; ++b) {
    float kb = kbA[b], nb = nbA[b], tb = (float)tgt[b];
    float p = fminf(fmaxf(tb / nb, EPS_), 1.0f - EPS_);
    float lp = kb * logf(p) + (nb - kb) * log1pf(-p)
             + lgammaf(nb + 1.0f) - lgammaf(kb + 1.0f) - lgammaf(nb - kb + 1.0f);
    loss += -lp / nb;
    sumk += kb;
    sumn += nb;
  }
  out3[0] = loss / (float)B_;   // loss
  out3[1] = sumk;               // num_boundaries
  out3[2] = sumn;               // total_positions
}

// ---------------------------------------------------------------------------
extern "C" void kernel_launch(void* const* d_in, const int* in_sizes, int n_in,
                              void* d_out, int out_size, void* d_ws, size_t ws_size,
                              hipStream_t stream) {
  (void)in_sizes; (void)n_in; (void)out_size; (void)ws_size;
  const float* hidden = (const float*)d_in[0];   // [B,L,D]
  const float* mask   = (const float*)d_in[1];   // [B,L]
  const float* Wq     = (const float*)d_in[2];   // [D,D]
  const float* Wk     = (const float*)d_in[3];   // [D,D]
  const int*   tgt    = (const int*)d_in[4];     // [B]

  float* out     = (float*)d_out;
  float* pooled  = out;                                   // B*L*D
  float* out3    = out + (size_t)B_ * L_ * D_;            // loss, num_b, total_pos
  float* sam     = out3 + 3;                              // B*L

  // workspace layout
  char* ws = (char*)d_ws;
  size_t off = 0;
  float* rnorm  = (float*)(ws + off); off += (size_t)NROWS_ * 4;            // 64KB
  float* WqT    = (float*)(ws + off); off += (size_t)D_ * D_ * 4;           // 4MB
  float* WkT    = (float*)(ws + off); off += (size_t)D_ * D_ * 4;           // 4MB
  float* qbuf   = (float*)(ws + off); off += (size_t)NROWS_ * D_ * 4;       // 64MB
  float* kbuf   = (float*)(ws + off); off += (size_t)NROWS_ * D_ * 4;       // 64MB
  float* cosbuf = (float*)(ws + off); off += (size_t)B_ * (L_ - 1) * 4;
  float* hardWS = (float*)(ws + off); off += (size_t)B_ * L_ * 4;
  int*   starts = (int*)(ws + off);   off += (size_t)B_ * (L_ + 1) * 4;
  int*   nsegA  = (int*)(ws + off);   off += 64;
  float* kbA    = (float*)(ws + off); off += 64;
  float* nbA    = (float*)(ws + off); off += 64;

  transpose1024_kernel<<<(D_ * D_) / 256, 256, 0, stream>>>(Wq, WqT);
  transpose1024_kernel<<<(D_ * D_) / 256, 256, 0, stream>>>(Wk, WkT);
  rnorm_kernel<<<NROWS_, 256, 0, stream>>>(hidden, rnorm);

  dim3 gemm_grid(NROWS_ / 16, D_ / 256);   // (1024, 4)
  gemm_wmma_kernel<<<gemm_grid, 128, 0, stream>>>(hidden, WqT, qbuf);
  gemm_wmma_kernel<<<gemm_grid, 128, 0, stream>>>(hidden, WkT, kbuf);

  cossim_kernel<<<(B_ * (L_ - 1) + 7) / 8, 256, 0, stream>>>(qbuf, kbuf, rnorm, cosbuf);

  boundary_kernel<<<B_, 256, 0, stream>>>(mask, cosbuf, hardWS, starts,
                                          nsegA, kbA, nbA, sam);

  dim3 pool_grid(L_, B_);
  pool_kernel<<<pool_grid, 256, 0, stream>>>(hidden, starts, nsegA, pooled);

  finalize_kernel<<<1, 32, 0, stream>>>(kbA, nbA, tgt, out3);
}